// MultiStageFIRFilter_25297357373554
// MI455X (gfx1250) — compile-verified
//
#include <hip/hip_runtime.h>

typedef float v2f __attribute__((ext_vector_type(2)));
typedef float v8f __attribute__((ext_vector_type(8)));

#define BATCH    8
#define NFRM     500
#define FP       80           // frame period
#define T_TOT    (NFRM * FP)  // 40000
#define TAPS     25
#define STAGES   20
#define CHUNK    2000
#define NCHUNK   (T_TOT / CHUNK)   // 20
#define HALO     560               // multiple of 80; > 27*19 + 27 = 540
#define XLEN     (HALO + CHUNK + 16)
#define MCSTRIDE 28                // 25 taps padded to 28 (7 chunks of 4)
#define AROWS    3                 // row0 = mc[n], row1 = mc[n+1]-mc[n], row2 = zeros
#define AFRAME   (AROWS * MCSTRIDE)  // 84 floats per frame
#define MCROWS   34

// One workgroup handles one (batch, chunk). All 20 Taylor stages run in LDS.
__global__ __launch_bounds__(256) void mlsa_fir_kernel(
    const float* __restrict__ x,   // (B, T)
    const float* __restrict__ mc,  // (B, N, 25)
    float* __restrict__ out)       // (B, T)
{
    __shared__ float xbuf[2][XLEN];
    __shared__ float ybuf[CHUNK];
    __shared__ float amat[MCROWS * AFRAME];   // per-frame WMMA A rows

    const int chunk    = blockIdx.x;
    const int b        = blockIdx.y;
    const int start    = chunk * CHUNK;
    const int end      = start + CHUNK;
    const int buf_base = start - HALO;          // negative for chunk 0
    const int tid      = threadIdx.x;

    // ---- init: zero both ping-pong buffers (so every read is finite) ----
    for (int i = tid; i < XLEN; i += 256) { xbuf[0][i] = 0.0f; xbuf[1][i] = 0.0f; }

    // ---- load x chunk + halo (zeros for t < 0) ----
    const float* xb = x + (size_t)b * T_TOT;
    for (int i = tid; i < HALO + CHUNK; i += 256) {
        int t = buf_base + i;
        if (t >= 0) xbuf[0][i] = xb[t];
    }
    // ---- y starts as x (owned region) ----
    for (int i = tid; i < CHUNK; i += 256) ybuf[i] = xb[start + i];

    // ---- build per-frame A-matrix rows: [row0 = mc[n] | row1 = dmc | row2 = 0] ----
    const int bb   = buf_base > 0 ? buf_base : 0;
    const int nlo  = bb / FP;
    const int nhi2 = (end - 1) / FP + 1;
    const int NF   = nhi2 - nlo + 1;            // <= 33
    for (int i = tid; i < NF * AFRAME; i += 256) {
        int f   = nlo + i / AFRAME;
        int rem = i % AFRAME;
        int r   = rem / MCSTRIDE;
        int j   = rem % MCSTRIDE;
        float v = 0.0f;
        if (j < TAPS && r < 2) {
            int fc  = f < (NFRM - 1) ? f : (NFRM - 1);
            int fc1 = (f + 1) < (NFRM - 1) ? (f + 1) : (NFRM - 1);
            float m0 = mc[((size_t)b * NFRM + fc) * TAPS + j];
            v = (r == 0) ? m0 : (mc[((size_t)b * NFRM + fc1) * TAPS + j] - m0);
        }
        amat[i] = v;
    }
    __syncthreads();

    const int lane = tid & 31;
    const int wave = tid >> 5;   // 8 waves
    const int m    = lane & 15;  // A row / B,D column
    const int hi   = lane >> 4;  // lane half holds K pair {0,1} vs {2,3}
    const int koff = hi << 1;
    const int arow = (m < 2 ? m : 2) * MCSTRIDE + koff;  // branchless A row select

    for (int a = 1; a <= STAGES; ++a) {
        const float* __restrict__ xin  = xbuf[(a - 1) & 1];
        float* __restrict__       xout = xbuf[a & 1];
        const int   Sa   = start - 27 * (STAGES - a);  // store threshold this stage
        const int   i0   = (Sa > 0 ? Sa : 0) >> 4;     // absolute 16-aligned tiles
        const int   iEnd = end >> 4;
        const float inva = 1.0f / (float)a;

        for (int i = i0 + wave; i < iEnd; i += 8) {
            const int t0 = i << 4;                     // tile base, inside one frame
            const int fr = t0 / FP - nlo;
            const float* __restrict__ Abase = &amat[fr * AFRAME + arow];

            v8f acc = {};
            const int tB = t0 + m - koff - buf_base;   // LDS index of B.x at chunk 0
#pragma unroll
            for (int c = 0; c < 7; ++c) {
                // A: one aligned 8-byte LDS load (rows 2..15 hit the zero row)
                v2f A = *(const v2f*)(Abase + 4 * c);
                // B: column n holds x[t0 + n - (4c+koff)] / x[.. - 1]
                const int li = tB - 4 * c;
                v2f B = {xin[li], xin[li - 1]};
                acc = __builtin_amdgcn_wmma_f32_16x16x4_f32(
                    false, A, false, B, (short)0, acc, false, false);
            }
            // D[0,n] = conv(mc[n]) at t0+n  -> acc[0], lanes 0..15
            // D[1,n] = conv(dmc)   at t0+n  -> acc[1], lanes 0..15
            const int t = t0 + m;
            if (hi == 0 && t >= Sa && t < end) {
                float w  = (float)(t % FP) * (1.0f / (float)FP);
                float xn = (acc[0] + w * acc[1]) * inva;
                xout[t - buf_base] = xn;
                if (t >= start) ybuf[t - start] += xn;
            }
        }
        __syncthreads();
    }

    float* ob = out + (size_t)b * T_TOT + start;
    for (int i = tid; i < CHUNK; i += 256) ob[i] = ybuf[i];
}

extern "C" void kernel_launch(void* const* d_in, const int* in_sizes, int n_in,
                              void* d_out, int out_size, void* d_ws, size_t ws_size,
                              hipStream_t stream) {
    (void)in_sizes; (void)n_in; (void)d_ws; (void)ws_size; (void)out_size;
    const float* x  = (const float*)d_in[0];   // (8, 40000) f32
    const float* mc = (const float*)d_in[1];   // (8, 500, 25) f32
    float* out      = (float*)d_out;           // (8, 40000) f32
    dim3 grid(NCHUNK, BATCH);                  // 20 x 8 = 160 workgroups
    mlsa_fir_kernel<<<grid, 256, 0, stream>>>(x, mc, out);
}